// SymplecticNet_84009560310421
// MI455X (gfx1250) — compile-verified
//
#include <hip/hip_runtime.h>
#include <stdint.h>

// SymplecticNet on MI455X (gfx1250) — single fused kernel.
//
// N=2, DIM=62 redundant monomial features (degree 1..5). Each 62-weight
// vector collapses exactly to 20 bivariate coefficients c_{a,b}; the gradient
// of features(x)@w is two degree-<=4 polynomials over the 15 shared monomials
// x0^a x1^b (a+b<=4). Per block:
//   1) async-stage raw weights (4 x 1984 floats) into LDS
//      (gfx1250 GLOBAL_LOAD_ASYNC_TO_LDS + s_wait_asynccnt),
//   2) threads 0..127 collapse the 128 weight vectors -> 3840-float gradient
//      coefficient table in LDS (fully unrolled, register-resident),
//   3) 32 symplectic sub-steps per sample, all state in VGPRs, coefficients
//      via broadcast ds loads.
// Roofline: ~0.46 GFLOP fp32 vs ~2 MB HBM -> VALU/latency bound; WMMA shapes
// (16x16 tiles) cannot help a 2-wide output with per-lane monomials.

#define MACRO_STEPS 32
#define DIMW 62
#define WTOT (MACRO_STEPS * DIMW)            // 1984 floats per weight array
#define RAW_TOTAL (4 * WTOT)                 // 7936 floats
#define COEF_PER_POLY 30                     // g0[15] + g1[15]
#define COEF_PER_STEP 120                    // 4 polys (A,B,C,D)
#define COEF_TOTAL (MACRO_STEPS * COEF_PER_STEP)   // 3840 floats
#define LDS_FLOATS (RAW_TOTAL + COEF_TOTAL)  // 11776 floats = 47104 B

// ---------------------------------------------------------------------------
// Gradient of collapsed polynomial at (x0,x1): 15 shared monomials, two
// 15-term fma chains. c points into LDS (wave-uniform -> broadcast ds loads).
// ---------------------------------------------------------------------------
__device__ __forceinline__ void grad15(const float* __restrict__ c,
                                       float x0, float x1,
                                       float& g0o, float& g1o) {
  float m3  = x0 * x0;
  float m4  = x0 * x1;
  float m5  = x1 * x1;
  float m6  = m3 * x0;
  float m7  = m3 * x1;
  float m8  = x0 * m5;
  float m9  = m5 * x1;
  float m10 = m3 * m3;
  float m11 = m6 * x1;
  float m12 = m3 * m5;
  float m13 = x0 * m9;
  float m14 = m5 * m5;

  float g0 = c[0];
  g0 = fmaf(c[1],  x0,  g0);
  g0 = fmaf(c[2],  x1,  g0);
  g0 = fmaf(c[3],  m3,  g0);
  g0 = fmaf(c[4],  m4,  g0);
  g0 = fmaf(c[5],  m5,  g0);
  g0 = fmaf(c[6],  m6,  g0);
  g0 = fmaf(c[7],  m7,  g0);
  g0 = fmaf(c[8],  m8,  g0);
  g0 = fmaf(c[9],  m9,  g0);
  g0 = fmaf(c[10], m10, g0);
  g0 = fmaf(c[11], m11, g0);
  g0 = fmaf(c[12], m12, g0);
  g0 = fmaf(c[13], m13, g0);
  g0 = fmaf(c[14], m14, g0);

  const float* d = c + 15;
  float g1 = d[0];
  g1 = fmaf(d[1],  x0,  g1);
  g1 = fmaf(d[2],  x1,  g1);
  g1 = fmaf(d[3],  m3,  g1);
  g1 = fmaf(d[4],  m4,  g1);
  g1 = fmaf(d[5],  m5,  g1);
  g1 = fmaf(d[6],  m6,  g1);
  g1 = fmaf(d[7],  m7,  g1);
  g1 = fmaf(d[8],  m8,  g1);
  g1 = fmaf(d[9],  m9,  g1);
  g1 = fmaf(d[10], m10, g1);
  g1 = fmaf(d[11], m11, g1);
  g1 = fmaf(d[12], m12, g1);
  g1 = fmaf(d[13], m13, g1);
  g1 = fmaf(d[14], m14, g1);

  g0o = g0;
  g1o = g1;
}

// ---------------------------------------------------------------------------
__global__ __launch_bounds__(256)
void symp_flow_kernel(const float4* __restrict__ zin,
                      const float* __restrict__ Aw,
                      const float* __restrict__ Bw,
                      const float* __restrict__ Cw,
                      const float* __restrict__ Dw,
                      const int* __restrict__ nsp,
                      float4* __restrict__ zout,
                      int batch) {
  __shared__ float lds[LDS_FLOATS];  // [0,7936): raw weights; [7936,+3840): coeffs

  const int tid = threadIdx.x;

  // ---- phase 1: async-stage raw weights into LDS (gfx1250 async ops) ------
  {
    // Generic pointers to LDS carry the 32-bit LDS offset in their low bits
    // (ISA 10.2 aperture mapping) == the addrspace(3) pointer value.
    uint32_t lraw = (uint32_t)(uintptr_t)(&lds[0]);
    const float* srcs[4] = {Aw, Bw, Cw, Dw};
#pragma unroll
    for (int a4 = 0; a4 < 4; ++a4) {
      uint64_t gbase = (uint64_t)(uintptr_t)srcs[a4];
      uint32_t lofs0 = lraw + (uint32_t)(a4 * WTOT) * 4u;
#pragma unroll
      for (int it = 0; it < (WTOT + 255) / 256; ++it) {
        uint32_t idx = (uint32_t)tid + (uint32_t)it * 256u;
        if (idx < (uint32_t)WTOT) {  // EXEC-predicated async load (per-lane)
          uint32_t loff = lofs0 + idx * 4u;  // VDST: per-lane LDS dest
          uint32_t goff = idx * 4u;          // VADDR: per-lane offset (GVS)
          asm volatile("global_load_async_to_lds_b32 %0, %1, %2"
                       :
                       : "v"(loff), "v"(goff), "s"(gbase)
                       : "memory");
        }
      }
    }
#if __has_builtin(__builtin_amdgcn_s_wait_asynccnt)
    __builtin_amdgcn_s_wait_asynccnt(0);
#else
    asm volatile("s_wait_asynccnt 0" ::: "memory");
#endif
  }
  __syncthreads();

  // ---- phase 2: collapse 128 weight vectors -> gradient coefficients ------
  // c_{d,b} = sum over degree-d index tuples m (bits of m = indices) with
  // popcount(m)==b of w[off_d+m]. g0 coeff(a,b) = (a+1)*c_{a+1,b};
  // g1 coeff(a,b) = (b+1)*c_{a,b+1}.
  if (tid < 4 * MACRO_STEPS) {
    int step = tid >> 2;
    int arr  = tid & 3;
    const float* w = &lds[arr * WTOT + step * DIMW];

    float c[6][6];
#pragma unroll
    for (int d = 0; d < 6; ++d)
#pragma unroll
      for (int b = 0; b < 6; ++b) c[d][b] = 0.0f;

    const int off[6] = {0, 0, 2, 6, 14, 30};
#pragma unroll
    for (int d = 1; d <= 5; ++d) {
#pragma unroll
      for (int m = 0; m < (1 << 5); ++m) {
        if (m < (1 << d)) c[d][__popc(m)] += w[off[d] + m];
      }
    }

    // monomial order k -> (a,b): degree-major, then b ascending
    const int MA[15] = {0,1,0,2,1,0,3,2,1,0,4,3,2,1,0};
    const int MB[15] = {0,0,1,0,1,2,0,1,2,3,0,1,2,3,4};

    float* out = &lds[RAW_TOTAL + (step * 4 + arr) * COEF_PER_POLY];
#pragma unroll
    for (int k = 0; k < 15; ++k) {
      int a = MA[k], b = MB[k];
      out[k]      = (float)(a + 1) * c[a + 1 + b][b];      // d/dx0 coeff
      out[15 + k] = (float)(b + 1) * c[a + b + 1][b + 1];  // d/dx1 coeff
    }
  }
  __syncthreads();

  // ---- phase 3: 32 symplectic sub-steps, state in VGPRs -------------------
  int i = blockIdx.x * blockDim.x + tid;
  if (i >= batch) return;

  float4 z = zin[i];
  float q0 = z.x, q1 = z.y, p0 = z.z, p1 = z.w;

  int ns = *nsp;  // wave-uniform scalar
  float dt  = (float)(1.0 / (32.0 * (double)ns));
  float hdt = 0.5f * dt;

  const float* ctab = &lds[RAW_TOTAL];
  for (int s = 0; s < MACRO_STEPS; ++s) {
    const float* cs = &ctab[s * COEF_PER_STEP];

    // drift A (half), kick B, drift A (half)
    float dA0, dA1;
    grad15(cs + 0, p0, p1, dA0, dA1);
    q0 = fmaf(dA0, hdt, q0);
    q1 = fmaf(dA1, hdt, q1);
    float dB0, dB1;
    grad15(cs + 30, q0, q1, dB0, dB1);
    p0 = fmaf(-dB0, dt, p0);
    p1 = fmaf(-dB1, dt, p1);
    q0 = fmaf(dA0, hdt, q0);
    q1 = fmaf(dA1, hdt, q1);

    // rotated pair: u = (q1, p0), v = (q0, p1)
    float u0 = q1, u1 = p0, v0 = q0, v1 = p1;
    float dC0, dC1;
    grad15(cs + 60, v0, v1, dC0, dC1);   // dCmod = (dC1, -dC0)
    u0 = fmaf(dC1, hdt, u0);
    u1 = fmaf(-dC0, hdt, u1);
    float dD0, dD1;
    grad15(cs + 90, u0, u1, dD0, dD1);   // dDmod = (dD1, -dD0)
    v0 = fmaf(dD1, dt, v0);
    v1 = fmaf(-dD0, dt, v1);
    u0 = fmaf(dC1, hdt, u0);
    u1 = fmaf(-dC0, hdt, u1);

    // z = [v0, u0, u1, v1]
    q0 = v0; q1 = u0; p0 = u1; p1 = v1;
  }

  zout[i] = make_float4(q0, q1, p0, p1);
}

// ---------------------------------------------------------------------------
extern "C" void kernel_launch(void* const* d_in, const int* in_sizes, int n_in,
                              void* d_out, int out_size, void* d_ws, size_t ws_size,
                              hipStream_t stream) {
  (void)d_ws; (void)ws_size; (void)n_in; (void)out_size;

  const float* z  = (const float*)d_in[0];
  const float* Aw = (const float*)d_in[1];
  const float* Bw = (const float*)d_in[2];
  const float* Cw = (const float*)d_in[3];
  const float* Dw = (const float*)d_in[4];
  const int*   ns = (const int*)d_in[5];

  int batch  = in_sizes[0] / 4;
  int blocks = (batch + 255) / 256;
  symp_flow_kernel<<<blocks, 256, 0, stream>>>(
      (const float4*)z, Aw, Bw, Cw, Dw, ns, (float4*)d_out, batch);
}